// BatchSRU_27556510171508
// MI455X (gfx1250) — compile-verified
//
#include <hip/hip_runtime.h>

typedef float v2f __attribute__((ext_vector_type(2)));
typedef float v4f __attribute__((ext_vector_type(4)));
typedef float v8f __attribute__((ext_vector_type(8)));

namespace {
constexpr int L  = 2048;
constexpr int Bn = 8;
constexpr int D  = 128;
constexpr int NB = 16;
constexpr int M  = L * Bn;        // 16384 rows per instance (l*Bn + b)
constexpr int H  = 3 * D;         // 384 output columns per instance
constexpr int S  = 16;            // scan segments
constexpr int SEG = L / S;        // 128 timesteps per segment
constexpr int CH = Bn * NB * D;   // 16384 independent recurrence channels
}

__device__ __forceinline__ float sigmoidf_(float v) {
  return 1.0f / (1.0f + __expf(-v));
}

// ---------------------------------------------------------------------------
// Kernel 1: U[k][m][h] = X_k[m][:] @ W_k[:][h], fused bias+sigmoid on gate
// columns (h >= 128). f32 WMMA 16x16x4. Each wave computes a 32(M)x96(N)
// strip: 2 M-tiles sharing one set of B fragments -> 12 WMMA per 16 loads.
// grid = (M/32, NB), block = 128 (4 wave32, each owning a 96-col N strip)
// ---------------------------------------------------------------------------
__global__ void __launch_bounds__(128)
sru_gemm(const float* __restrict__ x, const float* __restrict__ W,
         const float* __restrict__ bias, float* __restrict__ U) {
  const int tid  = threadIdx.x;
  const int lane = tid & 31;
  const int wv   = tid >> 5;        // wave 0..3 -> N strip
  const int half = lane >> 4;       // 0: K pair {0,1} / rows 0-7, 1: K pair {2,3} / rows 8-15
  const int lrow = lane & 15;
  const int k    = blockIdx.y;      // SRU instance
  const int m0   = blockIdx.x * 32; // double M tile

  // A fragment bases: lane holds row m0(+16)+lrow, K = kb + 2*half + {0,1}
  // x element (m, d, k) lives at x[(m*D + d)*NB + k]
  const float* px0 = x + ((size_t)(m0 + lrow) * D + 2 * half) * NB + k;
  const float* px1 = px0 + (size_t)16 * D * NB;

  // B fragment bases: lane holds column n, K = kb + 2*half + {0,1}
  const int n0w = wv * 96;
  const float* pw[6];
#pragma unroll
  for (int t = 0; t < 6; ++t) {
    const int n = n0w + t * 16 + lrow;
    pw[t] = W + (size_t)k * D * H + (size_t)(2 * half) * H + n;
  }

  v8f acc[12];                      // acc[t]   -> m-tile 0, col tile t
#pragma unroll                      // acc[6+t] -> m-tile 1, col tile t
  for (int t = 0; t < 12; ++t) acc[t] = {};

  for (int kb = 0; kb < D; kb += 4) {
    v2f a0, a1;
    a0.x = px0[(size_t)kb * NB];
    a0.y = px0[(size_t)kb * NB + NB];
    a1.x = px1[(size_t)kb * NB];
    a1.y = px1[(size_t)kb * NB + NB];
    v2f bfrag[6];
#pragma unroll
    for (int t = 0; t < 6; ++t) {
      bfrag[t].x = pw[t][(size_t)kb * H];
      bfrag[t].y = pw[t][(size_t)kb * H + H];
    }
#pragma unroll
    for (int t = 0; t < 6; ++t) {
      acc[t] = __builtin_amdgcn_wmma_f32_16x16x4_f32(
          false, a0, false, bfrag[t], (short)0, acc[t], false, false);
      acc[6 + t] = __builtin_amdgcn_wmma_f32_16x16x4_f32(
          false, a1, false, bfrag[t], (short)0, acc[6 + t], false, false);
    }
  }

  // Epilogue: C layout -> row = r + 8*half, col = lrow. Gates get bias+sigmoid.
#pragma unroll
  for (int i = 0; i < 2; ++i) {
#pragma unroll
    for (int t = 0; t < 6; ++t) {
      const int n = n0w + t * 16 + lrow;
      const float bval = (n >= D) ? bias[k * 2 * D + (n - D)] : 0.0f;
#pragma unroll
      for (int r = 0; r < 8; ++r) {
        const int m = m0 + 16 * i + r + 8 * half;
        float v = acc[6 * i + t][r];
        if (n >= D) v = sigmoidf_(v + bval);
        U[((size_t)k * M + m) * H + n] = v;
      }
    }
  }
}

// ---------------------------------------------------------------------------
// Kernel 2: per-segment affine reduction. c_out = P*c_in + Q with
// P = prod(f_t), Q = scan result starting from 0.
// grid = (Bn*NB, S), block = 32; each lane owns 4 consecutive d channels so
// all U traffic moves as b128 loads.
// ---------------------------------------------------------------------------
__global__ void __launch_bounds__(32)
sru_seg_pq(const float* __restrict__ U, float* __restrict__ Parr,
           float* __restrict__ Qarr) {
  const int d0 = threadIdx.x * 4;
  const int bk = blockIdx.x;          // b*NB + k
  const int s  = blockIdx.y;
  const int b_ = bk / NB;
  const int k  = bk % NB;
  const float* Ub = U + (size_t)k * M * H;

  v4f P = {1.0f, 1.0f, 1.0f, 1.0f};
  v4f Q = {};
#pragma unroll 4
  for (int t = 0; t < SEG; ++t) {
    const int m = (s * SEG + t) * Bn + b_;
    const size_t o = (size_t)m * H;
    const v4f xt = *(const v4f*)(Ub + o + d0);
    const v4f f  = *(const v4f*)(Ub + o + D + d0);
    P *= f;
    Q = f * Q + (1.0f - f) * xt;
  }
  const int ch = bk * D + d0;
  *(v4f*)(Parr + (size_t)s * CH + ch) = P;
  *(v4f*)(Qarr + (size_t)s * CH + ch) = Q;
}

// ---------------------------------------------------------------------------
// Kernel 3: tiny sequential scan over the 16 segment carries per channel.
// grid = CH/256, block = 256
// ---------------------------------------------------------------------------
__global__ void __launch_bounds__(256)
sru_seg_scan(const float* __restrict__ Parr, const float* __restrict__ Qarr,
             float* __restrict__ Cin) {
  const int ch = blockIdx.x * 256 + threadIdx.x;
  float c = 0.0f;
#pragma unroll
  for (int s = 0; s < S; ++s) {
    Cin[(size_t)s * CH + ch] = c;
    c = Parr[(size_t)s * CH + ch] * c + Qarr[(size_t)s * CH + ch];
  }
}

// ---------------------------------------------------------------------------
// Kernel 4: replay each segment with the correct carry-in, fuse highway
// combine and write output in the original (L,B,D,NB) layout.
// grid = (Bn*NB, S), block = 32; U read as b128, x/out stay scalar (stride NB).
// ---------------------------------------------------------------------------
__global__ void __launch_bounds__(32)
sru_apply(const float* __restrict__ U, const float* __restrict__ x,
          const float* __restrict__ Cin, float* __restrict__ out) {
  const int d0 = threadIdx.x * 4;
  const int bk = blockIdx.x;
  const int s  = blockIdx.y;
  const int b_ = bk / NB;
  const int k  = bk % NB;
  const float* Ub = U + (size_t)k * M * H;

  v4f c = *(const v4f*)(Cin + (size_t)s * CH + bk * D + d0);
#pragma unroll 4
  for (int t = 0; t < SEG; ++t) {
    const int m = (s * SEG + t) * Bn + b_;
    const size_t o = (size_t)m * H;
    const v4f xt = *(const v4f*)(Ub + o + d0);
    const v4f f  = *(const v4f*)(Ub + o + D + d0);
    const v4f r  = *(const v4f*)(Ub + o + 2 * D + d0);
    c = f * c + (1.0f - f) * xt;
    const size_t xo = ((size_t)m * D + d0) * NB + k;
#pragma unroll
    for (int j = 0; j < 4; ++j) {
      const float xin = x[xo + (size_t)j * NB];
      out[xo + (size_t)j * NB] = r[j] * c[j] + (1.0f - r[j]) * xin;
    }
  }
}

// ---------------------------------------------------------------------------
extern "C" void kernel_launch(void* const* d_in, const int* in_sizes, int n_in,
                              void* d_out, int out_size, void* d_ws, size_t ws_size,
                              hipStream_t stream) {
  (void)in_sizes; (void)n_in; (void)out_size; (void)ws_size;
  const float* x = (const float*)d_in[0];
  const float* W = (const float*)d_in[1];
  const float* b = (const float*)d_in[2];
  float* out = (float*)d_out;

  // workspace layout
  float* U    = (float*)d_ws;                 // NB*M*H floats (402 MB)
  float* Parr = U + (size_t)NB * M * H;       // S*CH floats
  float* Qarr = Parr + (size_t)S * CH;        // S*CH floats
  float* Cin  = Qarr + (size_t)S * CH;        // S*CH floats

  dim3 g1(M / 32, NB);
  sru_gemm<<<g1, 128, 0, stream>>>(x, W, b, U);

  dim3 g2(Bn * NB, S);
  sru_seg_pq<<<g2, 32, 0, stream>>>(U, Parr, Qarr);

  sru_seg_scan<<<CH / 256, 256, 0, stream>>>(Parr, Qarr, Cin);

  sru_apply<<<g2, 32, 0, stream>>>(U, x, Cin, out);
}